// GATNet_30210799960818
// MI455X (gfx1250) — compile-verified
//
#include <hip/hip_runtime.h>
#include <hip/hip_bf16.h>

typedef __attribute__((ext_vector_type(2))) float v2f;
typedef __attribute__((ext_vector_type(8))) float v8f;

#define NEG_SLOPE 0.2f

// ---------------------------------------------------------------------------
// GEMM via V_WMMA_F32_16X16X4_F32 (exact FP32 path).
// One wave computes one 16x16 tile of H = A[N,K] @ W[K,NC].
// A layout (16x4 f32): lanes 0-15 -> M=lane, VGPR0/1 = K=0/1; lanes 16-31 -> K=2/3.
// B layout (4x16 f32): VGPR0: K=0 (lanes 0-15), K=2 (lanes 16-31); VGPR1: K=1/K=3.
// D layout: VGPR v -> M=v (lanes 0-15) / M=v+8 (lanes 16-31), col = lane&15.
// ---------------------------------------------------------------------------
template <int KDIM, int NC, int CT>
__global__ void gemm_wmma_kernel(const float* __restrict__ A,
                                 const float* __restrict__ W,
                                 float* __restrict__ H, int nRowTiles) {
  int wave = (blockIdx.x * blockDim.x + threadIdx.x) >> 5;
  int lane = threadIdx.x & 31;
  if (wave >= nRowTiles * CT) return;  // wave-uniform guard (EXEC stays full)

  int rt = wave / CT;
  int ct = wave - rt * CT;
  int r = rt << 4;
  int lrow = lane & 15;
  int koff = (lane >> 4) << 1;  // 0 or 2

  int col = (ct << 4) + lrow;
  bool colok = (col < NC);
  int colc = colok ? col : 0;

  const float* arow = A + (size_t)(r + lrow) * KDIM;
  v8f acc = {};
#pragma unroll
  for (int k = 0; k < KDIM; k += 4) {
    v2f a, b;
    a.x = arow[k + koff];
    a.y = arow[k + koff + 1];
    float b0 = W[(k + koff) * NC + colc];
    float b1 = W[(k + koff + 1) * NC + colc];
    b.x = colok ? b0 : 0.0f;
    b.y = colok ? b1 : 0.0f;
    acc = __builtin_amdgcn_wmma_f32_16x16x4_f32(false, a, false, b,
                                                (short)0, acc, false, false);
  }
  if (colok) {
    int rbase = r + ((lane >> 4) << 3);
#pragma unroll
    for (int v = 0; v < 8; ++v)
      H[(size_t)(rbase + v) * NC + col] = acc[v];
  }
}

// alpha_s/alpha_d: per (node,head) dot of h[n,head,:] with a_src/a_dst[head,:]
template <int TH, int TOD>
__global__ void alpha_kernel(const float* __restrict__ h,
                             const float* __restrict__ a_src,
                             const float* __restrict__ a_dst,
                             float* __restrict__ as, float* __restrict__ ad,
                             int total /* = N*TH */) {
  int i = blockIdx.x * blockDim.x + threadIdx.x;
  if (i >= total) return;
  int head = i % TH;
  const float* hp = h + (size_t)i * TOD;  // (i/TH)*TH*TOD + head*TOD == i*TOD
  float s = 0.f, d = 0.f;
#pragma unroll
  for (int o = 0; o < TOD; ++o) {
    float v = hp[o];
    s += v * a_src[head * TOD + o];
    d += v * a_dst[head * TOD + o];
  }
  as[i] = s;
  ad[i] = d;
}

__device__ __forceinline__ void edge_nodes(const int* __restrict__ ei, int E,
                                           int e, int& s, int& d) {
  if (e < E) { s = ei[e]; d = ei[E + e]; }
  else       { s = e - E; d = e - E; }   // self loops appended
}

__device__ __forceinline__ unsigned f2key(float x) {
  unsigned u = __float_as_uint(x);
  return (u & 0x80000000u) ? ~u : (u | 0x80000000u);
}
__device__ __forceinline__ float key2f(unsigned k) {
  return __uint_as_float((k & 0x80000000u) ? (k & 0x7FFFFFFFu) : ~k);
}

// Pass A: segment max of leaky-relu edge logits via uint-key atomicMax
template <int TH>
__global__ void edge_max_kernel(const int* __restrict__ ei, int E, int ET,
                                const float* __restrict__ as,
                                const float* __restrict__ ad,
                                unsigned* __restrict__ mkey) {
  int idx = blockIdx.x * blockDim.x + threadIdx.x;
  if (idx >= ET * TH) return;
  int e = idx / TH, h = idx - (idx / TH) * TH;
  int s, d;
  edge_nodes(ei, E, e, s, d);
  float x = as[s * TH + h] + ad[d * TH + h];
  x = (x > 0.f) ? x : NEG_SLOPE * x;
  atomicMax(&mkey[d * TH + h], f2key(x));
}

// Pass B: denom = segment_sum(exp(e - m))
template <int TH>
__global__ void edge_sum_kernel(const int* __restrict__ ei, int E, int ET,
                                const float* __restrict__ as,
                                const float* __restrict__ ad,
                                const unsigned* __restrict__ mkey,
                                float* __restrict__ denom) {
  int idx = blockIdx.x * blockDim.x + threadIdx.x;
  if (idx >= ET * TH) return;
  int e = idx / TH, h = idx - (idx / TH) * TH;
  int s, d;
  edge_nodes(ei, E, e, s, d);
  float x = as[s * TH + h] + ad[d * TH + h];
  x = (x > 0.f) ? x : NEG_SLOPE * x;
  float m = key2f(mkey[d * TH + h]);
  atomicAdd(&denom[d * TH + h], __expf(x - m));
}

// Pass C: agg[dst, col] += h[src, col] * softmax_coef(edge, head(col))
template <int TH, int TOD>
__global__ void edge_agg_kernel(const int* __restrict__ ei, int E, int ET,
                                const float* __restrict__ hmat,
                                const float* __restrict__ as,
                                const float* __restrict__ ad,
                                const unsigned* __restrict__ mkey,
                                const float* __restrict__ denom,
                                float* __restrict__ agg) {
  const int HD = TH * TOD;
  int idx = blockIdx.x * blockDim.x + threadIdx.x;
  if (idx >= ET * HD) return;
  int e = idx / HD;
  int col = idx - e * HD;
  int h = col / TOD;
  int s, d;
  edge_nodes(ei, E, e, s, d);
  float x = as[s * TH + h] + ad[d * TH + h];
  x = (x > 0.f) ? x : NEG_SLOPE * x;
  float m = key2f(mkey[d * TH + h]);
  float coef = __expf(x - m) / denom[d * TH + h];
  atomicAdd(&agg[(size_t)d * HD + col], hmat[(size_t)s * HD + col] * coef);
}

__global__ void zero_kernel(float* __restrict__ p, int n) {
  int i = blockIdx.x * blockDim.x + threadIdx.x;
  if (i < n) p[i] = 0.0f;
}

// layer-1 epilogue: x = elu(agg + b), in place, 64 cols
__global__ void elu_bias64_kernel(float* __restrict__ x,
                                  const float* __restrict__ b, int n) {
  int i = blockIdx.x * blockDim.x + threadIdx.x;
  if (i >= n) return;
  float v = x[i] + b[i & 63];
  x[i] = (v > 0.f) ? v : (__expf(v) - 1.0f);
}

// final: in-place row-wise log_softmax over 40 cols (+bias); one wave per row
__global__ void logsoftmax40_kernel(float* __restrict__ out,
                                    const float* __restrict__ b, int N) {
  int wave = (blockIdx.x * blockDim.x + threadIdx.x) >> 5;
  int lane = threadIdx.x & 31;
  if (wave >= N) return;  // wave-uniform
  float* row = out + (size_t)wave * 40;
  float v0 = row[lane] + b[lane];
  float v1 = (lane < 8) ? (row[lane + 32] + b[lane + 32]) : -INFINITY;
  float mx = fmaxf(v0, v1);
#pragma unroll
  for (int off = 16; off; off >>= 1) mx = fmaxf(mx, __shfl_xor(mx, off, 32));
  float s = __expf(v0 - mx) + ((lane < 8) ? __expf(v1 - mx) : 0.f);
#pragma unroll
  for (int off = 16; off; off >>= 1) s += __shfl_xor(s, off, 32);
  float lse = mx + __logf(s);
  row[lane] = v0 - lse;
  if (lane < 8) row[lane + 32] = v1 - lse;
}

static inline int nb(long long n, int bs) { return (int)((n + bs - 1) / bs); }

extern "C" void kernel_launch(void* const* d_in, const int* in_sizes, int n_in,
                              void* d_out, int out_size, void* d_ws,
                              size_t ws_size, hipStream_t stream) {
  const float* x      = (const float*)d_in[0];
  const int*   ei     = (const int*)d_in[1];
  const float* W1     = (const float*)d_in[2];
  const float* a1s_p  = (const float*)d_in[3];
  const float* a1d_p  = (const float*)d_in[4];
  const float* b1     = (const float*)d_in[5];
  const float* W2     = (const float*)d_in[6];
  const float* a2s_p  = (const float*)d_in[7];
  const float* a2d_p  = (const float*)d_in[8];
  const float* b2     = (const float*)d_in[9];

  const int N  = in_sizes[0] / 128;  // 50000
  const int E  = in_sizes[1] / 2;    // 800000
  const int ET = E + N;              // + self loops
  const int RT = (N + 15) / 16;      // 3125 row tiles (exact)

  // workspace carve-up (floats), ~41 MB total
  float* ws = (float*)d_ws;
  size_t o = 0;
  float*    h1    = ws + o; o += (size_t)N * 64;
  float*    a1s   = ws + o; o += (size_t)N * 8;
  float*    a1d   = ws + o; o += (size_t)N * 8;
  unsigned* mkey1 = (unsigned*)(ws + o); o += (size_t)N * 8;
  float*    den1  = ws + o; o += (size_t)N * 8;   // adjacent to mkey1
  float*    x2    = ws + o; o += (size_t)N * 64;  // agg1 -> elu -> layer2 input
  float*    h2    = ws + o; o += (size_t)N * 40;
  float*    a2s   = ws + o; o += (size_t)N;
  float*    a2d   = ws + o; o += (size_t)N;
  unsigned* mkey2 = (unsigned*)(ws + o); o += (size_t)N;
  float*    den2  = ws + o; o += (size_t)N;       // adjacent to mkey2
  float*    agg2  = (float*)d_out;                // N x 40 accumulator

  const int BS = 256;

  // ---- Layer 1 ----
  gemm_wmma_kernel<128, 64, 4><<<nb((long long)RT * 4 * 32, BS), BS, 0, stream>>>(
      x, W1, h1, RT);
  alpha_kernel<8, 8><<<nb((long long)N * 8, BS), BS, 0, stream>>>(
      h1, a1s_p, a1d_p, a1s, a1d, N * 8);
  zero_kernel<<<nb((long long)N * 16, BS), BS, 0, stream>>>((float*)mkey1, N * 16);
  edge_max_kernel<8><<<nb((long long)ET * 8, BS), BS, 0, stream>>>(
      ei, E, ET, a1s, a1d, mkey1);
  edge_sum_kernel<8><<<nb((long long)ET * 8, BS), BS, 0, stream>>>(
      ei, E, ET, a1s, a1d, mkey1, den1);
  zero_kernel<<<nb((long long)N * 64, BS), BS, 0, stream>>>(x2, N * 64);
  edge_agg_kernel<8, 8><<<nb((long long)ET * 64, BS), BS, 0, stream>>>(
      ei, E, ET, h1, a1s, a1d, mkey1, den1, x2);
  elu_bias64_kernel<<<nb((long long)N * 64, BS), BS, 0, stream>>>(x2, b1, N * 64);

  // ---- Layer 2 ----
  gemm_wmma_kernel<64, 40, 3><<<nb((long long)RT * 3 * 32, BS), BS, 0, stream>>>(
      x2, W2, h2, RT);
  alpha_kernel<1, 40><<<nb((long long)N, BS), BS, 0, stream>>>(
      h2, a2s_p, a2d_p, a2s, a2d, N);
  zero_kernel<<<nb((long long)N * 2, BS), BS, 0, stream>>>((float*)mkey2, N * 2);
  edge_max_kernel<1><<<nb((long long)ET, BS), BS, 0, stream>>>(
      ei, E, ET, a2s, a2d, mkey2);
  edge_sum_kernel<1><<<nb((long long)ET, BS), BS, 0, stream>>>(
      ei, E, ET, a2s, a2d, mkey2, den2);
  zero_kernel<<<nb((long long)N * 40, BS), BS, 0, stream>>>(agg2, N * 40);
  edge_agg_kernel<1, 40><<<nb((long long)ET * 40, BS), BS, 0, stream>>>(
      ei, E, ET, h2, a2s, a2d, mkey2, den2, agg2);
  logsoftmax40_kernel<<<nb((long long)N * 32, BS), BS, 0, stream>>>(agg2, b2, N);
}